// LAPContrastiveLossWithHardMining_51067161150313
// MI455X (gfx1250) — compile-verified
//
#include <hip/hip_runtime.h>
#include <hip/hip_bf16.h>

typedef __attribute__((ext_vector_type(2))) float v2f;
typedef __attribute__((ext_vector_type(8))) float v8f;

#define KCHUNKS 4      // K-dimension split: 4x more waves for latency hiding
#define DDIM    1024   // fixed by the reference problem

// ---------------------------------------------------------------------------
// Kernel 1: row-wise dot products via V_WMMA_F32_16X16X4_F32 diagonal trick.
// One wave32 owns a (16-row tile, K-chunk) pair. A = q rows (16 x Kc),
// B = d rows as columns (Kc x 16); diag(A.B) = per-row partial dots.
//
// Fragment layouts (CDNA5 ISA 7.12.2):
//   A 16x4 f32 : lanes 0-15 -> M=lane, VGPR0=K0,VGPR1=K1 ; lanes 16-31 -> K2,K3
//   B  4x16 f32: lanes 0-15 -> N=lane, VGPR0=K0,VGPR1=K1 ; lanes 16-31 -> K2,K3
// => identical per-lane address for A and B fragments:
//      row = base + (lane&15), col = k + ((lane>>4)*2), 2 consecutive floats.
//   C/D 16x16 f32: VGPR r, lanes 0-15 -> M=r, N=lane ; lanes 16-31 -> M=r+8.
//
// Memory pipeline: 4 explicitly named register buffers (distinct live ranges
// the allocator cannot merge) so ~36-48 b64 loads stay outstanding while the
// WMMAs consume a retired buffer. Round-2 codegen showed that a plain
// unrolled loop collapses to ONE register triple (3 loads in flight, full
// latency stall per group) -- this structure prevents that.
// ---------------------------------------------------------------------------

// one buffer = 4 k-steps = 16 K-values = 12 b64 loads (24 VGPRs)
#define LAP_DECL(buf) v2f buf##a[4], buf##p[4], buf##n[4]
#define LAP_LOAD(buf, koff)                                   \
  do {                                                        \
    _Pragma("unroll")                                         \
    for (int j = 0; j < 4; ++j) {                             \
      buf##a[j] = *(const v2f*)(qp + (koff) + 4 * j);         \
      buf##p[j] = *(const v2f*)(pp + (koff) + 4 * j);         \
      buf##n[j] = *(const v2f*)(np + (koff) + 4 * j);         \
    }                                                         \
  } while (0)
#define LAP_CONS(buf)                                                         \
  do {                                                                        \
    _Pragma("unroll")                                                         \
    for (int j = 0; j < 4; ++j) {                                             \
      accp = __builtin_amdgcn_wmma_f32_16x16x4_f32(false, buf##a[j], false,   \
                 buf##p[j], (short)0, accp, false, false);                    \
      accn = __builtin_amdgcn_wmma_f32_16x16x4_f32(false, buf##a[j], false,   \
                 buf##n[j], (short)0, accn, false, false);                    \
    }                                                                         \
  } while (0)

__global__ void __launch_bounds__(256)
lap_rowdot_wmma(const float* __restrict__ q,
                const float* __restrict__ dpos,
                const float* __restrict__ dneg,
                float* __restrict__ part,
                int B) {
  const int tid   = blockIdx.x * blockDim.x + threadIdx.x;
  const int wave  = tid >> 5;
  const int lane  = tid & 31;
  const int chunk = wave & (KCHUNKS - 1);
  const int tile  = wave >> 2;               // log2(KCHUNKS) == 2
  const int rowBase = tile << 4;
  const int kLen = DDIM / KCHUNKS;           // 256 = 16 batches of 16
  const int k0   = chunk * kLen;

  const int m  = lane & 15;                  // row (A) / col (B) inside tile
  const int kk = (lane >> 4) << 1;           // lanes 16-31 hold K2,K3

  const size_t rowOff = (size_t)(rowBase + m) * (size_t)DDIM + (size_t)(k0 + kk);
  const float* qp = q    + rowOff;
  const float* pp = dpos + rowOff;
  const float* np = dneg + rowOff;

  v8f accp = {};
  v8f accn = {};

  LAP_DECL(b0); LAP_DECL(b1); LAP_DECL(b2); LAP_DECL(b3);

  // prologue: fill all 4 buffers (48 loads in flight)
  LAP_LOAD(b0, 0);
  LAP_LOAD(b1, 16);
  LAP_LOAD(b2, 32);
  LAP_LOAD(b3, 48);

  // steady state: consume one buffer, immediately refill it
  for (int k = 0; k + 64 < kLen; k += 64) {   // k = 0, 64, 128
    LAP_CONS(b0); LAP_LOAD(b0, k + 64);
    LAP_CONS(b1); LAP_LOAD(b1, k + 80);
    LAP_CONS(b2); LAP_LOAD(b2, k + 96);
    LAP_CONS(b3); LAP_LOAD(b3, k + 112);
  }
  // epilogue: drain last 4 buffers (K = 192..255)
  LAP_CONS(b0); LAP_CONS(b1); LAP_CONS(b2); LAP_CONS(b3);

  // Diagonal element m sits at (vgpr m, lane m) for m<8 and at
  // (vgpr m-8, lane m+16) for m>=8. Source lanes {0..7, 24..31} hold the
  // needed value in acc[lane & 7]; lanes 0-15 gather and store.
  float vp = accp[lane & 7];
  float vn = accn[lane & 7];
  const int src = (lane < 8) ? lane : (lane + 16);
  vp = __shfl(vp, src, 32);
  vn = __shfl(vn, src, 32);
  if (lane < 16) {
    part[(size_t)chunk * B + rowBase + lane]             = vp;
    part[(size_t)(KCHUNKS + chunk) * B + rowBase + lane] = vn;
  }
}

// ---------------------------------------------------------------------------
// Kernel 2: exact top-k + masked relu-mean, one 1024-thread workgroup.
// sim value = fixed-order sum of the KCHUNKS partials (deterministic, so the
// same key is recomputed identically in every pass; all passes hit L2).
// Monotone float->uint key; 4-pass MSB-first radix select (256-bin LDS
// histogram, integer atomics -> deterministic); exact tie handling at the
// k-th key; fixed-order tree reduction for the float sum.
// Both losses are: mean over top-k largest v of relu(v - margin), with
//   pos: v =  s, margin = +0.1
//   neg: v = -s, margin = -0.8   (relu(0.8 - s) == relu(v - (-0.8)))
// ---------------------------------------------------------------------------
__device__ __forceinline__ unsigned lap_fkey(float f) {
  unsigned u = __float_as_uint(f);
  return u ^ ((u >> 31) ? 0xFFFFFFFFu : 0x80000000u);
}
__device__ __forceinline__ float lap_kinv(unsigned key) {
  unsigned u = (key >> 31) ? (key ^ 0x80000000u) : ~key;
  return __uint_as_float(u);
}
__device__ __forceinline__ float lap_load_sim(const float* __restrict__ base,
                                              int B, int i) {
  float v = base[i];
#pragma unroll
  for (int c = 1; c < KCHUNKS; ++c) v += base[(size_t)c * B + i];
  return v;
}

__global__ void __launch_bounds__(1024)
lap_select_loss(const float* __restrict__ part,
                float* __restrict__ out,
                int n, int k) {
  __shared__ unsigned hist[256];
  __shared__ float    red[1024];
  __shared__ unsigned sh_prefix;
  __shared__ int      sh_cgt;
  __shared__ float    sh_res;

  const int t = threadIdx.x;
  float losses[2];

  for (int which = 0; which < 2; ++which) {
    const float* s    = part + (which ? (size_t)KCHUNKS * n : 0);
    const float scale = which ? -1.0f : 1.0f;
    const float marg  = which ? -0.8f : 0.1f;

    if (t == 0) { sh_prefix = 0u; sh_cgt = 0; }
    __syncthreads();

    // ---- 4-pass radix select of the k-th largest key ----
    for (int pass = 0; pass < 4; ++pass) {
      const int shift = 24 - 8 * pass;
      const unsigned mask_hi = (pass == 0) ? 0u : ~((1u << (shift + 8)) - 1u);

      for (int i = t; i < 256; i += blockDim.x) hist[i] = 0u;
      __syncthreads();

      const unsigned prefix = sh_prefix;
      for (int i = t; i < n; i += blockDim.x) {
        unsigned key = lap_fkey(scale * lap_load_sim(s, n, i));
        if ((key & mask_hi) == prefix)
          atomicAdd(&hist[(key >> shift) & 255u], 1u);
      }
      __syncthreads();

      if (t == 0) {
        int rem = k - sh_cgt;     // ranks still to place within this prefix
        int d = 255;
        for (;; --d) {
          int c = (int)hist[d];
          if (c >= rem || d == 0) break;
          rem -= c;
        }
        sh_cgt    = k - rem;      // # keys strictly greater than new prefix
        sh_prefix = prefix | ((unsigned)d << shift);
      }
      __syncthreads();
    }

    const unsigned T  = sh_prefix;   // exact key of k-th largest value
    const int     cgt = sh_cgt;      // # strictly greater than T

    // ---- sum relu(v - margin) over values with key > T ----
    float psum = 0.0f;
    for (int i = t; i < n; i += blockDim.x) {
      float v = scale * lap_load_sim(s, n, i);
      if (lap_fkey(v) > T) psum += fmaxf(v - marg, 0.0f);
    }
    red[t] = psum;
    __syncthreads();
    for (int off = 512; off > 0; off >>= 1) {
      if (t < off) red[t] += red[t + off];
      __syncthreads();
    }
    if (t == 0) {
      float tv    = lap_kinv(T);
      float total = red[0] + (float)(k - cgt) * fmaxf(tv - marg, 0.0f);
      sh_res = total / (float)k;
    }
    __syncthreads();
    losses[which] = sh_res;
    __syncthreads();
  }

  if (t == 0) {
    out[0] = losses[0] + losses[1];  // total_loss
    out[1] = losses[0];              // loss_pos
    out[2] = losses[1];              // loss_neg
  }
}

// ---------------------------------------------------------------------------
extern "C" void kernel_launch(void* const* d_in, const int* in_sizes, int n_in,
                              void* d_out, int out_size, void* d_ws, size_t ws_size,
                              hipStream_t stream) {
  const float* q  = (const float*)d_in[0];
  const float* dp = (const float*)d_in[1];
  const float* dn = (const float*)d_in[2];
  float* out  = (float*)d_out;
  float* part = (float*)d_ws;         // 2 * KCHUNKS * B floats (512 KB)

  const int D = DDIM;
  const int B = in_sizes[0] / D;      // 16384

  // One wave per (16-row tile, K-chunk): (B/16)*KCHUNKS waves,
  // 8 waves (256 threads) per block.
  const int waves   = (B / 16) * KCHUNKS;   // 4096
  const int threads = waves * 32;           // 131072
  lap_rowdot_wmma<<<dim3(threads / 256), dim3(256), 0, stream>>>(
      q, dp, dn, part, B);

  int k = (B * 3) / 10;               // int(B * 0.3) == 4915 for B=16384
  if (k < 1) k = 1;
  lap_select_loss<<<dim3(1), dim3(1024), 0, stream>>>(part, out, B, k);
}